// dMaSIF_29858612642362
// MI455X (gfx1250) — compile-verified
//
#include <hip/hip_runtime.h>

typedef __attribute__((ext_vector_type(2))) float v2f;
typedef __attribute__((ext_vector_type(8))) float v8f;

__device__ __forceinline__ float leaky_f(float x) { return x >= 0.f ? x : 0.2f * x; }

// ---------------------------------------------------------------------------
// Pad one edge-MLP weight set: W1(33x33)->W1p(36x48), b1(33)->b1p(48),
// W2(33x16)->W2p(36x16). Zero fill so WMMA K/N tiles are exact.
// ---------------------------------------------------------------------------
__global__ void pad_weights_kernel(const float* __restrict__ W1,
                                   const float* __restrict__ b1,
                                   const float* __restrict__ W2,
                                   float* __restrict__ W1p,
                                   float* __restrict__ b1p,
                                   float* __restrict__ W2p) {
    int t = threadIdx.x;
    int stride = blockDim.x;
    for (int i = t; i < 36 * 48; i += stride) {
        int r = i / 48, c = i % 48;
        W1p[i] = (r < 33 && c < 33) ? W1[r * 33 + c] : 0.f;
    }
    for (int i = t; i < 48; i += stride) b1p[i] = (i < 33) ? b1[i] : 0.f;
    for (int i = t; i < 36 * 16; i += stride) {
        int r = i / 16, c = i % 16;
        W2p[i] = (r < 33) ? W2[r * 16 + c] : 0.f;
    }
}

__global__ void fill_ones_kernel(float* __restrict__ p, int n) {
    int i = blockIdx.x * blockDim.x + threadIdx.x;
    if (i < n) p[i] = 1.0f;
}

// ---------------------------------------------------------------------------
// Type MLP: Y = leaky(X @ W1 + b1) @ W2 + b2, all 16x16. One wave = 16 rows.
// Uses V_WMMA_F32_16X16X4_F32, K=16 -> 4 steps per GEMM.
// ---------------------------------------------------------------------------
#define TT_WAVES 8
__global__ __launch_bounds__(TT_WAVES * 32)
void type_mlp_kernel(const float* __restrict__ X, const float* __restrict__ W1,
                     const float* __restrict__ b1, const float* __restrict__ W2,
                     const float* __restrict__ b2, float* __restrict__ Y, int N) {
    __shared__ float lds[TT_WAVES][16][17];
    const int wave = threadIdx.x >> 5;
    const int lane = threadIdx.x & 31;
    const int base = (blockIdx.x * TT_WAVES + wave) * 16;
    if (base >= N) return;
    const int half = lane >> 4;
    const int l15 = lane & 15;
    const int koff = half * 2;
    float(*H)[17] = lds[wave];

    // GEMM1: X(16x16) @ W1(16x16)
    v8f acc = {};
#pragma unroll
    for (int kk = 0; kk < 4; ++kk) {
        const int kr = kk * 4 + koff;
        v2f a;
        a.x = X[(base + l15) * 16 + kr];
        a.y = X[(base + l15) * 16 + kr + 1];
        v2f b;
        b.x = W1[kr * 16 + l15];
        b.y = W1[(kr + 1) * 16 + l15];
        acc = __builtin_amdgcn_wmma_f32_16x16x4_f32(false, a, false, b, (short)0,
                                                    acc, false, false);
    }
    // bias + leaky -> LDS (row = atom, col = hidden dim)
    {
        const int mb = half * 8;
        const float bb = b1[l15];
#pragma unroll
        for (int v = 0; v < 8; ++v) H[mb + v][l15] = leaky_f(acc[v] + bb);
    }
    __builtin_amdgcn_wave_barrier();

    // GEMM2: H(16x16) @ W2(16x16)
    v8f acc2 = {};
#pragma unroll
    for (int kk = 0; kk < 4; ++kk) {
        const int kr = kk * 4 + koff;
        v2f a;
        a.x = H[l15][kr];
        a.y = H[l15][kr + 1];
        v2f b;
        b.x = W2[kr * 16 + l15];
        b.y = W2[(kr + 1) * 16 + l15];
        acc2 = __builtin_amdgcn_wmma_f32_16x16x4_f32(false, a, false, b, (short)0,
                                                     acc2, false, false);
    }
    const float bb2 = b2[l15];
#pragma unroll
    for (int v = 0; v < 8; ++v)
        Y[(base + v + half * 8) * 16 + l15] = acc2[v] + bb2;
}

// ---------------------------------------------------------------------------
// Brute-force KNN (k=16, squared L2, ascending). One thread per query,
// LDS-tiled candidate points, register insertion list.
// ---------------------------------------------------------------------------
__global__ __launch_bounds__(256)
void knn_kernel(const float* __restrict__ Q, int nq, const float* __restrict__ P,
                int np, int* __restrict__ oidx, float* __restrict__ odist,
                int exclude_self) {
    __shared__ float tile[256 * 3];
    const int tid = threadIdx.x;
    const int q = blockIdx.x * blockDim.x + tid;
    const bool qv = q < nq;
    float qx = 0.f, qy = 0.f, qz = 0.f;
    if (qv) { qx = Q[q * 3]; qy = Q[q * 3 + 1]; qz = Q[q * 3 + 2]; }
    float bd[16];
    int bi[16];
#pragma unroll
    for (int t = 0; t < 16; ++t) { bd[t] = 3.4e38f; bi[t] = 0; }

    for (int tb = 0; tb < np; tb += 256) {
        const int j = tb + tid;
        if (j < np) {
            tile[tid * 3] = P[j * 3];
            tile[tid * 3 + 1] = P[j * 3 + 1];
            tile[tid * 3 + 2] = P[j * 3 + 2];
        }
        __syncthreads();
        const int lim = min(256, np - tb);
        for (int u = 0; u < lim; ++u) {
            const int jj = tb + u;
            if (exclude_self && jj == q) continue;
            const float dx = qx - tile[u * 3];
            const float dy = qy - tile[u * 3 + 1];
            const float dz = qz - tile[u * 3 + 2];
            const float d = dx * dx + dy * dy + dz * dz;
            if (d < bd[15]) {
                float cd = d;
                int ci = jj;
#pragma unroll
                for (int t = 0; t < 16; ++t) {
                    if (cd < bd[t]) {
                        float td = bd[t]; int ti = bi[t];
                        bd[t] = cd; bi[t] = ci; cd = td; ci = ti;
                    }
                }
            }
        }
        __syncthreads();
    }
    if (qv) {
#pragma unroll
        for (int t = 0; t < 16; ++t) {
            oidx[q * 16 + t] = bi[t];
            odist[q * 16 + t] = bd[t];
        }
    }
}

// ---------------------------------------------------------------------------
// One message-passing layer. One wave = one point (16 edges).
//   feat(16x36) = [center(16) | nb(16) | dist | 0 0 0]
//   H = leaky(feat @ W1p + b1p)        : 27 WMMA f32 16x16x4
//   M = H @ W2p                        :  9 WMMA
//   msg = sum_edges(M) + 16*b2 ; out = Fc + leaky(groupnorm(msg))
// ---------------------------------------------------------------------------
#define EDGE_WAVES 8
#define HROW 49  // odd stride -> conflict-free across 64 banks
__global__ __launch_bounds__(EDGE_WAVES * 32)
void edge_layer_kernel(const float* __restrict__ Fc, const float* __restrict__ Fn,
                       const int* __restrict__ idx, const float* __restrict__ dist,
                       const float* __restrict__ W1p, const float* __restrict__ b1p,
                       const float* __restrict__ W2p, const float* __restrict__ b2,
                       const float* __restrict__ gamma, const float* __restrict__ beta,
                       float* __restrict__ Fout, int Np) {
    __shared__ float lds[EDGE_WAVES][16][HROW];
    const int wave = threadIdx.x >> 5;
    const int lane = threadIdx.x & 31;
    const int p = blockIdx.x * EDGE_WAVES + wave;
    if (p >= Np) return;  // whole-wave exit; EXEC stays all-ones for WMMA
    float(*F)[HROW] = lds[wave];

    const int half = lane >> 4;
    const int l15 = lane & 15;
    const int koff = half * 2;

    // ---- stage feat into LDS ----
    {   // neighbor features: each lane covers half of one edge row
        const int row = lane >> 1;
        const int cb = (lane & 1) * 8;
        const int nb = idx[p * 16 + row];
#pragma unroll
        for (int t = 0; t < 8; ++t)
            F[row][16 + cb + t] = Fn[nb * 16 + cb + t];
    }
    {   // broadcast center features to all 16 rows
        const float cv = Fc[p * 16 + l15];
        const int r0 = half * 8;
#pragma unroll
        for (int r = 0; r < 8; ++r) F[r0 + r][l15] = cv;
    }
    if (lane < 16) {  // distance + K-padding
        F[lane][32] = dist[p * 16 + lane];
        F[lane][33] = 0.f; F[lane][34] = 0.f; F[lane][35] = 0.f;
    }
    __builtin_amdgcn_wave_barrier();

    // ---- GEMM1: H = leaky(feat(16x36) @ W1p(36x48) + b1p) ----
    v8f acc0 = {}, acc1 = {}, acc2 = {};
#pragma unroll
    for (int kk = 0; kk < 9; ++kk) {
        const int kr = kk * 4 + koff;
        v2f a;  a.x = F[l15][kr];               a.y = F[l15][kr + 1];
        v2f b0; b0.x = W1p[kr * 48 + l15];      b0.y = W1p[(kr + 1) * 48 + l15];
        v2f b1v; b1v.x = W1p[kr * 48 + 16 + l15]; b1v.y = W1p[(kr + 1) * 48 + 16 + l15];
        v2f b2v; b2v.x = W1p[kr * 48 + 32 + l15]; b2v.y = W1p[(kr + 1) * 48 + 32 + l15];
        acc0 = __builtin_amdgcn_wmma_f32_16x16x4_f32(false, a, false, b0, (short)0, acc0, false, false);
        acc1 = __builtin_amdgcn_wmma_f32_16x16x4_f32(false, a, false, b1v, (short)0, acc1, false, false);
        acc2 = __builtin_amdgcn_wmma_f32_16x16x4_f32(false, a, false, b2v, (short)0, acc2, false, false);
    }
    __builtin_amdgcn_wave_barrier();
    {   // bias + leaky, write H back to LDS (overwrites feat; reads all done)
        const int mb = half * 8;
        const float bb0 = b1p[l15], bb1 = b1p[16 + l15], bb2 = b1p[32 + l15];
#pragma unroll
        for (int v = 0; v < 8; ++v) {
            F[mb + v][l15] = leaky_f(acc0[v] + bb0);
            F[mb + v][16 + l15] = leaky_f(acc1[v] + bb1);
            F[mb + v][32 + l15] = leaky_f(acc2[v] + bb2);
        }
    }
    __builtin_amdgcn_wave_barrier();

    // ---- GEMM2: M = H(16x36) @ W2p(36x16) ----
    v8f accm = {};
#pragma unroll
    for (int kk = 0; kk < 9; ++kk) {
        const int kr = kk * 4 + koff;
        v2f a; a.x = F[l15][kr];          a.y = F[l15][kr + 1];
        v2f b; b.x = W2p[kr * 16 + l15];  b.y = W2p[(kr + 1) * 16 + l15];
        accm = __builtin_amdgcn_wmma_f32_16x16x4_f32(false, a, false, b, (short)0, accm, false, false);
    }

    // ---- sum over 16 edges (rows of M) + 16*b2 ----
    float s = accm[0] + accm[1] + accm[2] + accm[3] + accm[4] + accm[5] + accm[6] + accm[7];
    s += __shfl_xor(s, 16, 32);  // combine M=0..7 with M=8..15
    const float msg = s + 16.f * b2[l15];

    // ---- group norm: 2 groups of 8 channels (lanes are channels) ----
    float mean = msg;
    mean += __shfl_xor(mean, 1, 32);
    mean += __shfl_xor(mean, 2, 32);
    mean += __shfl_xor(mean, 4, 32);
    mean *= 0.125f;
    const float xm = msg - mean;
    float vs = xm * xm;
    vs += __shfl_xor(vs, 1, 32);
    vs += __shfl_xor(vs, 2, 32);
    vs += __shfl_xor(vs, 4, 32);
    vs *= 0.125f;
    const float xn = xm * rsqrtf(vs + 1e-5f);
    const float y = leaky_f(xn * gamma[l15] + beta[l15]);
    const float outv = Fc[p * 16 + l15] + y;
    if (lane < 16) Fout[p * 16 + lane] = outv;
}

// ---------------------------------------------------------------------------
extern "C" void kernel_launch(void* const* d_in, const int* in_sizes, int n_in,
                              void* d_out, int out_size, void* d_ws, size_t ws_size,
                              hipStream_t stream) {
    const float* xyz = (const float*)d_in[0];
    const float* atom_xyz = (const float*)d_in[1];
    const float* atomtypes = (const float*)d_in[2];
    const float* tt_W1 = (const float*)d_in[5];
    const float* tt_b1 = (const float*)d_in[6];
    const float* tt_W2 = (const float*)d_in[7];
    const float* tt_b2 = (const float*)d_in[8];
    const float* aa_W1 = (const float*)d_in[9];
    const float* aa_b1 = (const float*)d_in[10];
    const float* aa_W2 = (const float*)d_in[11];
    const float* aa_b2 = (const float*)d_in[12];
    const float* aa_gamma = (const float*)d_in[13];
    const float* aa_beta = (const float*)d_in[14];
    const float* em_W1 = (const float*)d_in[15];
    const float* em_b1 = (const float*)d_in[16];
    const float* em_W2 = (const float*)d_in[17];
    const float* em_b2 = (const float*)d_in[18];
    const float* em_gamma = (const float*)d_in[19];
    const float* em_beta = (const float*)d_in[20];

    const int NA_ = in_sizes[1] / 3;   // 8000
    const int NP_ = in_sizes[0] / 3;   // 30000

    // workspace layout (floats, then ints)
    float* ws = (float*)d_ws;
    size_t off = 0;
    float* outA = ws + off; off += (size_t)NA_ * 16;
    float* outB = ws + off; off += (size_t)NA_ * 16;
    float* ones = ws + off; off += (size_t)NP_ * 16;
    float* embA = ws + off; off += (size_t)NP_ * 16;
    float* embB = ws + off; off += (size_t)NP_ * 16;
    float* distA = ws + off; off += (size_t)NA_ * 16;
    float* dist2 = ws + off; off += (size_t)NP_ * 16;
    float* wpad = ws + off; off += (size_t)6 * 2352;  // per set: 1728 W1p | 48 b1p | 576 W2p
    int* idxA = (int*)(ws + off); off += (size_t)NA_ * 16;
    int* idx2 = (int*)(ws + off); off += (size_t)NP_ * 16;

    // 1) pad edge-MLP weights (aa sets 0..2, em sets 3..5)
    for (int i = 0; i < 3; ++i) {
        float* sa = wpad + (size_t)i * 2352;
        pad_weights_kernel<<<1, 256, 0, stream>>>(aa_W1 + i * 33 * 33, aa_b1 + i * 33,
                                                  aa_W2 + i * 33 * 16, sa, sa + 1728, sa + 1776);
        float* se = wpad + (size_t)(3 + i) * 2352;
        pad_weights_kernel<<<1, 256, 0, stream>>>(em_W1 + i * 33 * 33, em_b1 + i * 33,
                                                  em_W2 + i * 33 * 16, se, se + 1728, se + 1776);
    }

    // 2) atom type MLP -> outA
    {
        int blocks = (NA_ + TT_WAVES * 16 - 1) / (TT_WAVES * 16);
        type_mlp_kernel<<<blocks, TT_WAVES * 32, 0, stream>>>(atomtypes, tt_W1, tt_b1,
                                                              tt_W2, tt_b2, outA, NA_);
    }

    // 3) atom-atom KNN (k=16, excluding self)
    knn_kernel<<<(NA_ + 255) / 256, 256, 0, stream>>>(atom_xyz, NA_, atom_xyz, NA_,
                                                      idxA, distA, 1);

    // 4) 3 atom message-passing layers (ping-pong outA/outB)
    {
        int blocks = (NA_ + EDGE_WAVES - 1) / EDGE_WAVES;
        const float* src = outA;
        float* dst = outB;
        for (int i = 0; i < 3; ++i) {
            const float* sa = wpad + (size_t)i * 2352;
            edge_layer_kernel<<<blocks, EDGE_WAVES * 32, 0, stream>>>(
                src, src, idxA, distA, sa, sa + 1728, sa + 1776,
                aa_b2 + i * 16, aa_gamma + i * 16, aa_beta + i * 16, dst, NA_);
            const float* t = src; src = dst; dst = (float*)t;
        }
        // after 3 layers, final atom features are in outB (A->B->A->B)
    }

    // 5) surface-atom KNN (k=16)
    knn_kernel<<<(NP_ + 255) / 256, 256, 0, stream>>>(xyz, NP_, atom_xyz, NA_,
                                                      idx2, dist2, 0);

    // 6) emb init = ones
    fill_ones_kernel<<<(NP_ * 16 + 255) / 256, 256, 0, stream>>>(ones, NP_ * 16);

    // 7) 3 surface message-passing layers; neighbors always from final atom feats
    {
        int blocks = (NP_ + EDGE_WAVES - 1) / EDGE_WAVES;
        const float* atomF = outB;
        const float* src = ones;
        float* dsts[3] = {embA, embB, (float*)d_out};
        for (int i = 0; i < 3; ++i) {
            const float* se = wpad + (size_t)(3 + i) * 2352;
            edge_layer_kernel<<<blocks, EDGE_WAVES * 32, 0, stream>>>(
                src, atomF, idx2, dist2, se, se + 1728, se + 1776,
                em_b2 + i * 16, em_gamma + i * 16, em_beta + i * 16, dsts[i], NP_);
            src = dsts[i];
        }
    }
}